// SpectralLinear_56813827392405
// MI455X (gfx1250) — compile-verified
//
#include <hip/hip_runtime.h>
#include <hip/hip_bf16.h>

#define NDIM   1024
#define BATCH  4096
#define NREFL  2048   // 1024 U reflections + 1024 V reflections

typedef __attribute__((ext_vector_type(2))) float v2f;
typedef __attribute__((ext_vector_type(8))) float v8f;

__device__ __forceinline__ float wave_reduce_sum(float v) {
#pragma unroll
    for (int off = 16; off > 0; off >>= 1)
        v += __shfl_xor(v, off, 32);
    return v;
}

// ---------------------------------------------------------------------------
// Kernel 1: collapse the 2048 sequential Householder reflections (+ sigma
// scaling) into a single 1024x1024 operator P, built row-resident in LDS.
// 32 blocks x 1024 threads; wave w of block b owns row r = b*32 + w of P.
// Inner loops are float4-vectorized (ds_load_b128/ds_store_b128): the serial
// chain is the critical path of the whole workload, so per-step instruction
// count matters more than anything else.
// ---------------------------------------------------------------------------
__global__ __launch_bounds__(1024)
void build_P_kernel(const float* __restrict__ U,
                    const float* __restrict__ V,
                    const float* __restrict__ p,
                    float* __restrict__ P) {
    extern __shared__ float smem[];
    float* Mrow  = smem;                  // [32][1024]  own rows of P (128 KB)
    float* uvec  = smem + 32 * NDIM;      // [1024]      staged u vector
    float* invnn = uvec + NDIM;           // [2048]      2 / (u.u) per row
    float* sig   = invnn + NREFL;         // [1024]      sigma diagonal

    const int tid  = threadIdx.x;
    const int wave = tid >> 5;
    const int lane = tid & 31;
    const int r    = blockIdx.x * 32 + wave;   // global row of P owned here
    float* myrow   = Mrow + wave * NDIM;

    float4*       rowv = (float4*)myrow;       // 4 KB-aligned
    const float4* uv   = (const float4*)uvec;  // 128 KB-aligned

    // Init own rows of P to identity.
#pragma unroll
    for (int k = lane; k < NDIM; k += 32)
        myrow[k] = (k == r) ? 1.0f : 0.0f;

    // sigma[k] = 0.9*(sigmoid(p[k]) - 0.5) + 0.55   (r=0.45, mean=0.55)
    {
        float pv = p[tid];
        float s  = 1.0f / (1.0f + __expf(-pv));
        sig[tid] = 0.9f * (s - 0.5f) + 0.55f;
    }

    // Precompute 2/(u.u) for all 2048 reflection vectors (one row per wave).
    for (int base = 0; base < NREFL; base += 32) {
        const int row = base + wave;
        const float* src = (row < NDIM) ? (U + (size_t)row * NDIM)
                                        : (V + (size_t)(row - NDIM) * NDIM);
        float acc = 0.0f;
        for (int k = lane; k < NDIM; k += 32) {
            const float u = src[k];
            acc += u * u;
        }
        acc = wave_reduce_sum(acc);
        if (lane == 0) invnn[row] = 2.0f / acc;
    }
    __syncthreads();

    // Sequential chain: U rows 0..1023, then sigma, then V rows 1023..0.
    for (int step = 0; step < NREFL; ++step) {
        int srow; const float* src; int nidx;
        if (step < NDIM) { srow = step;        src = U + (size_t)srow * NDIM; nidx = srow; }
        else             { srow = 2047 - step; src = V + (size_t)srow * NDIM; nidx = NDIM + srow; }

        // Stage u cooperatively (1 element / thread, coalesced).
        uvec[tid] = src[tid];
        __syncthreads();

        // Sigma column scaling exactly between the two chains (row-local).
        if (step == NDIM) {
            for (int k = lane; k < NDIM; k += 32)
                myrow[k] *= sig[k];
        }

        // Triangular: u has a zero prefix of length srow — skip it at
        // float4-loop granularity (128 floats).
        const int kb4 = (srow >> 2) & ~31;   // float4 index, multiple of 32

        float dot = 0.0f;
        for (int k4 = kb4 + lane; k4 < NDIM / 4; k4 += 32) {
            const float4 m = rowv[k4];
            const float4 u = uv[k4];
            dot += m.x * u.x + m.y * u.y + m.z * u.z + m.w * u.w;
        }
        dot = wave_reduce_sum(dot);                 // all lanes hold the sum
        const float alpha = dot * invnn[nidx];

        for (int k4 = kb4 + lane; k4 < NDIM / 4; k4 += 32) {
            float4 m = rowv[k4];
            const float4 u = uv[k4];
            m.x -= alpha * u.x;
            m.y -= alpha * u.y;
            m.z -= alpha * u.z;
            m.w -= alpha * u.w;
            rowv[k4] = m;
        }

        __syncthreads();   // protect uvec before next step overwrites it
    }

    // Write P (row-major [k][n]) to workspace, 16B per lane.
    float4* Pv = (float4*)(P + (size_t)r * NDIM);
    for (int k4 = lane; k4 < NDIM / 4; k4 += 32)
        Pv[k4] = rowv[k4];
}

// ---------------------------------------------------------------------------
// Kernel 2: out = x @ P + bias via V_WMMA_F32_16X16X4_F32.
// Each wave computes a 32x64 output tile = 2x4 grid of 16x16 fragments:
// 8 WMMAs per k-step reusing 2 A-frags and 4 B-frags (6 B/lane loaded per
// WMMA vs 16 B/lane unblocked -> ~2.7x less L2 traffic).
// A frag (16x4 f32): lanes 0-15 hold K={0,1}; lanes 16-31 hold K={2,3}.
// C/D (8 VGPR): VGPR j -> row +j (lanes 0-15) / +8+j (lanes 16-31), col = lane&15.
// ---------------------------------------------------------------------------
__global__ __launch_bounds__(256)
void gemm_wmma_kernel(const float* __restrict__ X,
                      const float* __restrict__ P,
                      const float* __restrict__ bias,
                      float* __restrict__ Out) {
    const int lane   = threadIdx.x & 31;
    const int waveId = threadIdx.x >> 5;
    const int wg     = blockIdx.x * 8 + waveId;      // 0 .. 2047
    const int tm     = wg >> 4;                      // 128 row tiles (32 rows)
    const int tn     = wg & 15;                      // 16 col tiles (64 cols)
    const int m0 = tm * 32;
    const int n0 = tn * 64;

    const int half = lane >> 4;        // 0: K pair {0,1}, 1: K pair {2,3}
    const int lr   = lane & 15;

    const float* arow0 = X + (size_t)(m0 + lr) * NDIM + 2 * half;       // 8B aligned
    const float* arow1 = arow0 + (size_t)16 * NDIM;
    const float* bbase = P + (size_t)(2 * half) * NDIM + n0 + lr;

    v8f acc[2][4];
#pragma unroll
    for (int mi = 0; mi < 2; ++mi)
#pragma unroll
        for (int ni = 0; ni < 4; ++ni)
            acc[mi][ni] = (v8f){};

    for (int k0 = 0; k0 < NDIM; k0 += 4) {
        v2f a[2];
        a[0] = *(const v2f*)(arow0 + k0);
        a[1] = *(const v2f*)(arow1 + k0);

        v2f b[4];
#pragma unroll
        for (int ni = 0; ni < 4; ++ni) {
            b[ni].x = bbase[(size_t)k0 * NDIM + ni * 16];
            b[ni].y = bbase[(size_t)k0 * NDIM + ni * 16 + NDIM];
        }

#pragma unroll
        for (int mi = 0; mi < 2; ++mi)
#pragma unroll
            for (int ni = 0; ni < 4; ++ni)
                acc[mi][ni] = __builtin_amdgcn_wmma_f32_16x16x4_f32(
                                  false, a[mi], false, b[ni],
                                  (short)0, acc[mi][ni], false, false);
    }

#pragma unroll
    for (int ni = 0; ni < 4; ++ni) {
        const float bv = bias[n0 + ni * 16 + lr];
#pragma unroll
        for (int mi = 0; mi < 2; ++mi) {
#pragma unroll
            for (int j = 0; j < 8; ++j) {
                Out[(size_t)(m0 + mi * 16 + j + 8 * half) * NDIM + n0 + ni * 16 + lr]
                    = acc[mi][ni][j] + bv;
            }
        }
    }
}

// ---------------------------------------------------------------------------
// Host launch: inputs are (x, U, V, p, bias), all float32.
// d_ws holds P (1024*1024*4 = 4 MB).
// ---------------------------------------------------------------------------
extern "C" void kernel_launch(void* const* d_in, const int* in_sizes, int n_in,
                              void* d_out, int out_size, void* d_ws, size_t ws_size,
                              hipStream_t stream) {
    const float* x    = (const float*)d_in[0];
    const float* U    = (const float*)d_in[1];
    const float* V    = (const float*)d_in[2];
    const float* p    = (const float*)d_in[3];
    const float* bias = (const float*)d_in[4];
    float* out = (float*)d_out;
    float* P   = (float*)d_ws;

    const size_t shmem = (size_t)(32 * NDIM + NDIM + NREFL + NDIM) * sizeof(float); // 144 KB

    build_P_kernel<<<dim3(32), dim3(1024), shmem, stream>>>(U, V, p, P);

    // 128 x 16 = 2048 wave-tiles, 8 waves per 256-thread block.
    gemm_wmma_kernel<<<dim3(256), dim3(256), 0, stream>>>(x, P, bias, out);
}